// MoEGate_77326591197231
// MI455X (gfx1250) — compile-verified
//
#include <hip/hip_runtime.h>
#include <hip/hip_bf16.h>

typedef __attribute__((ext_vector_type(2))) float v2f;
typedef __attribute__((ext_vector_type(8))) float v8f;

#define HDIM 4096
#define NEXP 8
#define WAVES_PER_BLOCK 4
#define TOKENS_PER_WAVE 16
#define TOKENS_PER_BLOCK (WAVES_PER_BLOCK * TOKENS_PER_WAVE)
#define BLOCK_THREADS (WAVES_PER_BLOCK * 32)

// Router gate: logits = hs @ gw^T via V_WMMA_F32_16X16X4_F32, then per-token
// softmax + top-2 + renormalize. One wave handles a 16-token x 16-expert tile
// (experts 8..15 are zero padding).
__global__ __launch_bounds__(BLOCK_THREADS) void moe_gate_kernel(
    const float* __restrict__ hs,   // [T, HDIM] flattened hidden states
    const float* __restrict__ gw,   // [NEXP, HDIM] gate weight
    float* __restrict__ wout,       // [T, 2] top-k weights
    int* __restrict__ eout,         // [T, 2] selected experts
    int T) {
  __shared__ float sLogits[TOKENS_PER_BLOCK * 16];

  const int tid  = threadIdx.x;
  const int wave = tid >> 5;
  const int lane = tid & 31;
  const int m    = lane & 15;  // A: token row in tile; B: expert column
  const int kh   = lane >> 4;  // K-half (0: K=0,1 ; 1: K=2,3)

  const int tokenBase = blockIdx.x * TOKENS_PER_BLOCK + wave * TOKENS_PER_WAVE;
  int tok = tokenBase + m;
  if (tok > T - 1) tok = T - 1;  // clamp (keeps EXEC full for WMMA)

  // A operand: hidden row for this token, starting at K offset 2*kh.
  const float* aPtr = hs + (size_t)tok * HDIM + 2 * kh;
  // B operand: gate row for expert n = m (zero-pad n >= NEXP via select).
  const bool nValid = (m < NEXP);
  const float* bPtr = gw + (size_t)(nValid ? m : 0) * HDIM + 2 * kh;

  v8f c = {};  // 16x16 f32 accumulator (8 VGPRs)

#pragma unroll 8
  for (int k = 0; k < HDIM; k += 4) {
    // hidden_states: single-use 256MiB stream -> non-temporal
    v2f a = __builtin_nontemporal_load((const v2f*)(aPtr + k));
    // gate_weight: 128KB, reused by every wave -> regular temporal load
    v2f b = *(const v2f*)(bPtr + k);
    if (!nValid) { b.x = 0.0f; b.y = 0.0f; }  // lowers to v_cndmask, EXEC stays full
    c = __builtin_amdgcn_wmma_f32_16x16x4_f32(false, a, false, b,
                                              (short)0, c, false, false);
  }

  // Spill logits to LDS: D[row][col], lane holds rows i + 8*kh, col = m.
#pragma unroll
  for (int i = 0; i < 8; ++i) {
    sLogits[(wave * TOKENS_PER_WAVE + i + 8 * kh) * 16 + m] = c[i];
  }
  __syncthreads();

  // One thread per token: softmax over 8 experts, top-2, renormalize.
  if (tid < TOKENS_PER_BLOCK) {
    const int g = blockIdx.x * TOKENS_PER_BLOCK + tid;
    if (g < T) {
      float l[NEXP];
#pragma unroll
      for (int j = 0; j < NEXP; ++j) l[j] = sLogits[tid * 16 + j];

      float mx = l[0];
#pragma unroll
      for (int j = 1; j < NEXP; ++j) mx = fmaxf(mx, l[j]);

      float p[NEXP];
#pragma unroll
      for (int j = 0; j < NEXP; ++j) p[j] = expf(l[j] - mx);
      // softmax denominator cancels in the top-2 renormalization, skip it.

      // top-1 (strict > scan: lowest index wins ties, matches jax top_k)
      int i1 = 0;
      float v1 = p[0];
#pragma unroll
      for (int j = 1; j < NEXP; ++j) {
        if (p[j] > v1) { v1 = p[j]; i1 = j; }
      }
      // top-2
      int i2 = -1;
      float v2 = -1.0f;
#pragma unroll
      for (int j = 0; j < NEXP; ++j) {
        if (j != i1 && p[j] > v2) { v2 = p[j]; i2 = j; }
      }

      const float inv = 1.0f / (v1 + v2);
      wout[2 * g + 0] = v1 * inv;
      wout[2 * g + 1] = v2 * inv;
      eout[2 * g + 0] = i1;
      eout[2 * g + 1] = i2;
    }
  }
}

extern "C" void kernel_launch(void* const* d_in, const int* in_sizes, int n_in,
                              void* d_out, int out_size, void* d_ws, size_t ws_size,
                              hipStream_t stream) {
  const float* hs = (const float*)d_in[0];  // [4,4096,4096] fp32
  const float* gw = (const float*)d_in[1];  // [8,4096] fp32

  const int T = in_sizes[0] / HDIM;  // 16384 tokens

  float* wout = (float*)d_out;                          // [T,2] fp32
  int*   eout = (int*)((float*)d_out + (size_t)2 * T);  // [T,2] int32

  const int grid = (T + TOKENS_PER_BLOCK - 1) / TOKENS_PER_BLOCK;  // 256
  moe_gate_kernel<<<grid, BLOCK_THREADS, 0, stream>>>(hs, gw, wout, eout, T);
}